// GraphSage_79714593014086
// MI455X (gfx1250) — compile-verified
//
#include <hip/hip_runtime.h>

typedef float v2f __attribute__((ext_vector_type(2)));
typedef float v8f __attribute__((ext_vector_type(8)));

#define NB 4096
#define NS 25
#define ND 32

__global__ __launch_bounds__(256)
void graphsage_fused(const float* __restrict__ src,      // [B,32]
                     const float* __restrict__ src_neg,  // [B,25,32]
                     const float* __restrict__ snn,      // [B,25,25,32]
                     const float* __restrict__ dst,
                     const float* __restrict__ dst_neg,
                     const float* __restrict__ dnn,
                     const float* __restrict__ w2,       // [64,32]
                     const float* __restrict__ w1,       // [64,32]
                     const float* __restrict__ dw,       // [64,1]
                     float* __restrict__ out)            // [B,1]
{
  const int b   = blockIdx.x;
  const int tid = threadIdx.x;
  const float inv_s = 1.0f / (float)NS;

  // Padded LDS to avoid bank conflicts on the WMMA fragment reads/writes.
  __shared__ float A[2][32][68];     // [side][row 0..31][k 0..63]; rows 25..31 unused
  __shared__ float W2s[64 * 32];
  __shared__ float W1s[64 * 32];
  __shared__ float DWs[64];
  __shared__ float AGG[2][32][36];   // hop-2 GEMM result per side
  __shared__ float SCF[2][64];       // hop-1 concat features per side
  __shared__ float X[64];            // [s, d] concat

  // Stage weights (hits L2 across the 4096 blocks).
  for (int i = tid; i < 2048; i += 256) { W2s[i] = w2[i]; W1s[i] = w1[i]; }
  if (tid < 64) DWs[tid] = dw[tid];

  const float* snn_b = snn     + (size_t)b * (NS * NS * ND);
  const float* dnn_b = dnn     + (size_t)b * (NS * NS * ND);
  const float* sn_b  = src_neg + (size_t)b * (NS * ND);
  const float* dn_b  = dst_neg + (size_t)b * (NS * ND);

  // ---- hop-2 neighbor mean over axis=1: A[side][j][32+d] = (1/S) * sum_i nn[i][j][d]
  // For fixed i the (j,d) plane is contiguous (800 floats): float4, fully coalesced.
  for (int v = tid; v < 2 * 200; v += 256) {
    const int side = v / 200;
    const int q    = v % 200;                 // float4 index into 25*32 plane
    const float* base = (side ? dnn_b : snn_b) + q * 4;
    float4 acc = make_float4(0.f, 0.f, 0.f, 0.f);
#pragma unroll
    for (int i = 0; i < NS; ++i) {
      float4 t = *reinterpret_cast<const float4*>(base + i * (NS * ND));
      acc.x += t.x; acc.y += t.y; acc.z += t.z; acc.w += t.w;
    }
    const int j = (q * 4) / ND;
    const int d = (q * 4) % ND;
    A[side][j][32 + d + 0] = acc.x * inv_s;
    A[side][j][32 + d + 1] = acc.y * inv_s;
    A[side][j][32 + d + 2] = acc.z * inv_s;
    A[side][j][32 + d + 3] = acc.w * inv_s;
  }
  // ---- self neighbor features into A[side][j][0..31]
  for (int v = tid; v < 2 * 200; v += 256) {
    const int side = v / 200;
    const int q    = v % 200;
    float4 t = *reinterpret_cast<const float4*>((side ? dn_b : sn_b) + q * 4);
    const int j = (q * 4) / ND;
    const int d = (q * 4) % ND;
    A[side][j][d + 0] = t.x;
    A[side][j][d + 1] = t.y;
    A[side][j][d + 2] = t.z;
    A[side][j][d + 3] = t.w;
  }
  __syncthreads();

  // ---- hop-2 GEMM: AGG[side] (32x32) = A[side] (32x64) @ W2 (64x32), f32 WMMA.
  // 8 waves -> 8 output tiles: side(2) x Mtile(2) x Ntile(2); 16 k-steps of K=4.
  {
    const int wave = tid >> 5;
    const int lane = tid & 31;
    const int side = wave >> 2;
    const int mt   = (wave >> 1) & 1;
    const int nt   = wave & 1;
    const int lo   = lane & 15;
    const int hi   = lane >> 4;

    v8f c = {0.f, 0.f, 0.f, 0.f, 0.f, 0.f, 0.f, 0.f};
#pragma unroll
    for (int k0 = 0; k0 < 64; k0 += 4) {
      const int ka = k0 + 2 * hi;   // A/B VGPR0 holds K={0,2}, VGPR1 K={1,3}
      v2f a, bw;
      a.x  = A[side][mt * 16 + lo][ka + 0];
      a.y  = A[side][mt * 16 + lo][ka + 1];
      bw.x = W2s[(ka + 0) * 32 + nt * 16 + lo];
      bw.y = W2s[(ka + 1) * 32 + nt * 16 + lo];
      c = __builtin_amdgcn_wmma_f32_16x16x4_f32(
              /*neg_a=*/false, a, /*neg_b=*/false, bw,
              /*c_mod=*/(short)0, c, /*reuse_a=*/false, /*reuse_b=*/false);
    }
    // C layout: VGPR r -> row r (lanes 0-15) / row r+8 (lanes 16-31), col = lane&15
#pragma unroll
    for (int r = 0; r < 8; ++r)
      AGG[side][mt * 16 + r + 8 * hi][nt * 16 + lo] = c[r];
  }
  __syncthreads();

  // ---- hop-1: mean over S rows of AGG, build scf = [self(32), mean(32)]
  if (tid < 64) {
    const int ss = tid >> 5;
    const int d  = tid & 31;
    float s = 0.f;
#pragma unroll
    for (int j = 0; j < NS; ++j) s += AGG[ss][j][d];
    SCF[ss][32 + d] = s * inv_s;
    SCF[ss][d]      = (ss ? dst : src)[(size_t)b * ND + d];
  }
  __syncthreads();

  // ---- hop-1 matmul: X[ss*32+d] = SCF[ss] (1x64) @ W1 (64x32)
  if (tid < 64) {
    const int ss = tid >> 5;
    const int d  = tid & 31;
    float acc = 0.f;
#pragma unroll
    for (int k = 0; k < 64; ++k) acc += SCF[ss][k] * W1s[k * 32 + d];
    X[ss * 32 + d] = acc;
  }
  __syncthreads();

  // ---- dense + ReLU
  if (tid == 0) {
    float acc = 0.f;
#pragma unroll
    for (int k = 0; k < 64; ++k) acc += X[k] * DWs[k];
    out[b] = fmaxf(acc, 0.f);
  }
}

extern "C" void kernel_launch(void* const* d_in, const int* in_sizes, int n_in,
                              void* d_out, int out_size, void* d_ws, size_t ws_size,
                              hipStream_t stream) {
  const float* src     = (const float*)d_in[0];
  const float* src_neg = (const float*)d_in[1];
  const float* snn     = (const float*)d_in[2];
  const float* dst     = (const float*)d_in[3];
  const float* dst_neg = (const float*)d_in[4];
  const float* dnn     = (const float*)d_in[5];
  const float* w2      = (const float*)d_in[6];
  const float* w1      = (const float*)d_in[7];
  const float* dw      = (const float*)d_in[8];
  float* out = (float*)d_out;

  graphsage_fused<<<NB, 256, 0, stream>>>(src, src_neg, snn, dst, dst_neg, dnn,
                                          w2, w1, dw, out);
}